// VectorQuantizer_43147241456162
// MI455X (gfx1250) — compile-verified
//
#include <hip/hip_runtime.h>
#include <hip/hip_bf16.h>
#include <stddef.h>

typedef __attribute__((ext_vector_type(2))) float v2f;
typedef __attribute__((ext_vector_type(8))) float v8f;

#define K_CODES   1024
#define D_DIM     256
#define N_ROWS    65536           // 16 * 4096
#define DECAY     0.99f
#define ONE_M_D   0.01f
#define EPS       1e-5f

// ---------------- workspace layout (floats) ----------------
#define WS_ETP    0               // 256*1024 swizzled codebook (B-operand image)
#define WS_ESQ    262144          // 1024
#define WS_CNT    263168          // 1024 cluster counts
#define WS_DW     264192          // 1024*256 segment sums of z
#define WS_LOSS   526336          // 1
#define WS_IDX    526337          // 65536 int indices

// ---------------- d_out layout (floats) --------------------
#define O_Q       0               // 16777216 quantized_st
#define O_IDX     16777216        // 65536 indices (as float)
#define O_LOSS    16842752        // 1
#define O_EMB     16842753        // 262144 new_embeddings
#define O_CS      17104897        // 1024 new_cs
#define O_EMAW    17105921        // 262144 new_ema_w

// Zero a float range.
__global__ void vq_zero_kernel(float* __restrict__ p, int n) {
    int i = blockIdx.x * 256 + threadIdx.x;
    if (i < n) p[i] = 0.0f;
}

// Build the B-operand global image ETP:
//   tile t (codes 16t..16t+15), pair p (k=2p,2p+1), lane j (code 16t+j):
//   ETP[f],  f = t*4096 + p*32 + j*2 + c   =  E[(16t+j)*256 + 2p+c]
// Staging a tile into LDS is then a contiguous 16 KB copy.
__global__ void vq_prep_kernel(const float* __restrict__ E,
                               float* __restrict__ ETP) {
    int f = blockIdx.x * 256 + threadIdx.x;        // over K*D = 262144
    int t  = f >> 12;
    int r  = f & 4095;
    int p  = r >> 5;
    int jj = r & 31;
    int j  = jj >> 1;
    int c  = jj & 1;
    ETP[f] = E[(t * 16 + j) * D_DIM + 2 * p + c];
}

// esq[n] = sum_d E[n][d]^2
__global__ void vq_esq_kernel(const float* __restrict__ E,
                              float* __restrict__ esq) {
    int n = blockIdx.x * 256 + threadIdx.x;        // 1024 threads
    if (n >= K_CODES) return;
    const float* row = E + n * D_DIM;
    float s = 0.0f;
    #pragma unroll 8
    for (int d = 0; d < D_DIM; ++d) { float v = row[d]; s += v * v; }
    esq[n] = s;
}

// Fused distance-GEMM + argmin. One wave: 16 rows of z vs all 1024 codes.
// score[m][n] = esq[n] - 2*dot(z_m,e_n)  (||z||^2 constant per row, dropped).
// A (16x256 of z) register-resident; B tiles double-buffered through LDS.
__global__ __launch_bounds__(256)
void vq_argmin_kernel(const float* __restrict__ z,
                      const float* __restrict__ ETP,
                      const float* __restrict__ esq,
                      int* __restrict__ idx_out,
                      float* __restrict__ idxf_out) {
    __shared__ __align__(16) float smB[2][4096];   // 2 x 16 KB B tiles
    __shared__ float smEsq[K_CODES];               // 4 KB

    const int tid     = threadIdx.x;
    const int lane    = tid & 31;
    const int wave    = tid >> 5;
    const int rowBase = (blockIdx.x * 8 + wave) << 4;   // 16 rows per wave
    const int l16     = lane & 15;
    const int hi      = (lane >> 4) & 1;                // 0: K=0,1  1: K=2,3
    const int m       = rowBase + l16;

    // ---- A operand: 16x256 tile of z, resident in 128 VGPRs, reused 64x ----
    v2f a[64];
    const float* zrow = z + (size_t)m * D_DIM + 2 * hi;  // 8B-aligned
    #pragma unroll
    for (int s = 0; s < 64; ++s)
        a[s] = *(const v2f*)(zrow + 4 * s);

    // ---- stage esq + first B tile ----
    #pragma unroll
    for (int i = 0; i < 4; ++i)
        smEsq[tid + 256 * i] = esq[tid + 256 * i];

    const float4* gB = (const float4*)ETP + tid;         // tile stride 1024 f4
    float4 r0 = gB[0], r1 = gB[256], r2 = gB[512], r3 = gB[768];
    {
        float4* sB = (float4*)&smB[0][0] + tid;
        sB[0] = r0; sB[256] = r1; sB[512] = r2; sB[768] = r3;
    }

    float best[8];
    int   bidx[8];
    #pragma unroll
    for (int q = 0; q < 8; ++q) { best[q] = 3.4e38f; bidx[q] = 0; }

    // ---- 64 codebook tiles of 16 codes, double-buffered ----
    for (int t = 0; t < 64; ++t) {
        __syncthreads();                                  // smB[t&1] ready
        if (t < 63) {                                     // prefetch next tile
            const float4* g = gB + (size_t)(t + 1) * 1024;
            r0 = g[0]; r1 = g[256]; r2 = g[512]; r3 = g[768];
        }

        // B per step s: pair index p = 2s+hi  ->  float2 at p*16 + l16
        const v2f* bp = (const v2f*)&smB[t & 1][0] + hi * 16 + l16;
        v8f acc = {};
        #pragma unroll
        for (int c = 0; c < 8; ++c) {                     // 8 chunks of 8
            v2f b[8];
            #pragma unroll
            for (int s2 = 0; s2 < 8; ++s2)
                b[s2] = bp[(8 * c + s2) * 32];            // ds_load_b64
            #pragma unroll
            for (int s2 = 0; s2 < 8; ++s2)
                acc = __builtin_amdgcn_wmma_f32_16x16x4_f32(
                    false, a[8 * c + s2], false, b[s2], (short)0, acc,
                    false, false);
        }

        const int n = (t << 4) + l16;
        const float eq = smEsq[n];
        #pragma unroll
        for (int q = 0; q < 8; ++q) {
            float sc = eq - 2.0f * acc[q];
            if (sc < best[q]) { best[q] = sc; bidx[q] = n; }
        }

        if (t < 63) {                                     // commit next tile
            float4* sB = (float4*)&smB[(t + 1) & 1][0] + tid;
            sB[0] = r0; sB[256] = r1; sB[512] = r2; sB[768] = r3;
        }
    }

    // ---- argmin across the 16 lanes holding one row (wave32 shfl) ----
    #pragma unroll
    for (int q = 0; q < 8; ++q) {
        #pragma unroll
        for (int off = 8; off >= 1; off >>= 1) {
            float ob = __shfl_xor(best[q], off, 32);
            int   oi = __shfl_xor(bidx[q], off, 32);
            if (ob < best[q] || (ob == best[q] && oi < bidx[q])) {
                best[q] = ob; bidx[q] = oi;
            }
        }
    }
    if (l16 == 0) {
        // lane 0 -> rows rowBase+0..7 ; lane 16 -> rows rowBase+8..15
        const int rb = rowBase + (hi ? 8 : 0);
        #pragma unroll
        for (int q = 0; q < 8; ++q) {
            idx_out[rb + q]  = bidx[q];
            idxf_out[rb + q] = (float)bidx[q];
        }
    }
}

// One block per row: gather quantized, loss partial, segment sums.
__global__ __launch_bounds__(256)
void vq_gather_kernel(const float* __restrict__ z,
                      const float* __restrict__ E,
                      const int* __restrict__ idx,
                      float* __restrict__ q_out,
                      float* __restrict__ counts,
                      float* __restrict__ dw,
                      float* __restrict__ lossAcc) {
    const int row = blockIdx.x;
    const int d   = threadIdx.x;
    const int k   = idx[row];
    const float e  = E[k * D_DIM + d];
    const float zv = z[(size_t)row * D_DIM + d];
    q_out[(size_t)row * D_DIM + d] = e;   // quantized_st == quantized
    atomicAdd(&dw[k * D_DIM + d], zv);    // segment_sum(z, idx)
    const float diff = zv - e;
    __shared__ float red[256];
    red[d] = diff * diff;
    __syncthreads();
    for (int s = 128; s > 0; s >>= 1) {
        if (d < s) red[d] += red[d + s];
        __syncthreads();
    }
    if (d == 0) {
        atomicAdd(lossAcc, red[0]);
        atomicAdd(&counts[k], 1.0f);
    }
}

// new_cs with Laplace smoothing + loss scalar. Single 1024-thread block.
__global__ __launch_bounds__(1024)
void vq_ema_cs_kernel(const float* __restrict__ ecs,
                      const float* __restrict__ counts,
                      const float* __restrict__ lossAcc,
                      float* __restrict__ new_cs_out,
                      float* __restrict__ loss_out) {
    const int k = threadIdx.x;
    float pre = ecs[k] * DECAY + ONE_M_D * counts[k];
    __shared__ float red[1024];
    red[k] = pre;
    __syncthreads();
    for (int s = 512; s > 0; s >>= 1) {
        if (k < s) red[k] += red[k + s];
        __syncthreads();
    }
    const float n = red[0];
    new_cs_out[k] = (pre + EPS) / (n + (float)K_CODES * EPS) * n;
    if (k == 0) {
        // loss = codebook + 0.25*commitment = 1.25 * mean((z-q)^2)
        loss_out[0] = lossAcc[0] * 1.25f / ((float)N_ROWS * (float)D_DIM);
    }
}

// new_ema_w and new_embeddings.
__global__ void vq_ema_w_kernel(const float* __restrict__ ema_w,
                                const float* __restrict__ dw,
                                const float* __restrict__ new_cs,
                                float* __restrict__ new_ema_w_out,
                                float* __restrict__ new_emb_out) {
    int i = blockIdx.x * 256 + threadIdx.x;        // over K*D
    float w = ema_w[i] * DECAY + ONE_M_D * dw[i];
    new_ema_w_out[i] = w;
    new_emb_out[i]   = w / new_cs[i >> 8];
}

extern "C" void kernel_launch(void* const* d_in, const int* in_sizes, int n_in,
                              void* d_out, int out_size, void* d_ws, size_t ws_size,
                              hipStream_t stream) {
    const float* z     = (const float*)d_in[0];   // (16,4096,256)
    const float* E     = (const float*)d_in[1];   // (1024,256)
    const float* ecs   = (const float*)d_in[2];   // (1024,)
    const float* ema_w = (const float*)d_in[3];   // (1024,256)
    (void)in_sizes; (void)n_in; (void)out_size; (void)ws_size;

    float* ws  = (float*)d_ws;
    float* out = (float*)d_out;

    float* ETP     = ws + WS_ETP;
    float* esq     = ws + WS_ESQ;
    float* counts  = ws + WS_CNT;
    float* dw      = ws + WS_DW;
    float* lossAcc = ws + WS_LOSS;
    int*   idx     = (int*)(ws + WS_IDX);

    // 0) zero atomic accumulators (counts, dw, lossAcc are contiguous)
    {
        int nz = K_CODES + K_CODES * D_DIM + 1;    // 263169
        vq_zero_kernel<<<(nz + 255) / 256, 256, 0, stream>>>(ws + WS_CNT, nz);
    }
    // 1) prep: swizzle codebook into B-operand image + code norms
    vq_prep_kernel<<<(K_CODES * D_DIM) / 256, 256, 0, stream>>>(E, ETP);
    vq_esq_kernel<<<K_CODES / 256, 256, 0, stream>>>(E, esq);
    // 2) fused distance GEMM + argmin: 4096 waves, 8 waves/block
    vq_argmin_kernel<<<N_ROWS / 16 / 8, 256, 0, stream>>>(
        z, ETP, esq, idx, out + O_IDX);
    // 3) gather + loss + segment sums
    vq_gather_kernel<<<N_ROWS, 256, 0, stream>>>(
        z, E, idx, out + O_Q, counts, dw, lossAcc);
    // 4) EMA cluster sizes + loss
    vq_ema_cs_kernel<<<1, 1024, 0, stream>>>(
        ecs, counts, lossAcc, out + O_CS, out + O_LOSS);
    // 5) EMA weights + new embeddings
    vq_ema_w_kernel<<<(K_CODES * D_DIM) / 256, 256, 0, stream>>>(
        ema_w, dw, out + O_CS, out + O_EMAW, out + O_EMB);
}